// PlasticLSTM_47742856462701
// MI455X (gfx1250) — compile-verified
//
#include <hip/hip_runtime.h>
#include <hip/hip_bf16.h>

#define T_STEPS 256
#define BATCH   32
#define IN_DIM  128
#define HID     256
#define G4H     1024
#define CLIPV   2.0f

typedef float  v2f  __attribute__((ext_vector_type(2)));
typedef float  v4f  __attribute__((ext_vector_type(4)));
typedef float  v8f  __attribute__((ext_vector_type(8)));

// ---------------------------------------------------------------------------
// Kernel 0: repack Wh [4H,H] -> WhTP [H][H][4] gate-interleaved:
//   WhTP[(j*H + k)*4 + g] = Wh[(g*H + k)*H + j]
// so the scan's inner loop fetches all 4 gate weights for (j,k) with one
// coalesced 16-byte global_load_b128 per thread.
// ---------------------------------------------------------------------------
__global__ void wh_repack_kernel(const float* __restrict__ Wh,
                                 float* __restrict__ WhTP) {
  int idx = blockIdx.x * blockDim.x + threadIdx.x;  // 0 .. 4H*H-1 (coalesced read)
  int g = idx >> 16;          // gate      (0..3)
  int k = (idx >> 8) & 255;   // out unit  (0..255)
  int j = idx & 255;          // h index   (0..255)
  WhTP[(size_t)((j << 8) + k) * 4 + g] = Wh[idx];
}

// ---------------------------------------------------------------------------
// Kernel 1: xf[T*B,4H] = x[T*B,I] @ Wx^T + bx via V_WMMA_F32_16X16X4_F32.
// One wave per 16x16 output tile, 4 waves per block. 32 WMMAs per tile.
// A (16x4 f32): lanes 0-15 -> M=0..15, VGPR0=K0,VGPR1=K1; lanes 16-31 -> K2,K3.
// C/D (16x16 f32): VGPR v, lane<16 -> (M=v, N=lane); lane>=16 -> (M=v+8).
// ---------------------------------------------------------------------------
__global__ __launch_bounds__(128)
void xf_gemm_wmma_kernel(const float* __restrict__ x,
                         const float* __restrict__ Wx,
                         const float* __restrict__ bx,
                         float* __restrict__ xf) {
  const int lane = threadIdx.x;            // 0..31
  const int wave = threadIdx.y;            // 0..3
  const int tile = blockIdx.x * 4 + wave;  // 0..32767
  const int mt = tile & 511;               // 512 tiles along M=8192
  const int nt = tile >> 9;                // 64 tiles along N=1024
  const int m0 = mt << 4;
  const int n0 = nt << 4;
  const int lm = lane & 15;
  const int hi = lane >> 4;                // which half-wave (K pair select)

  const int rowA = m0 + lm;                // x row (M index)
  const int colB = n0 + lm;                // Wx row == output col (N index)

  v8f c = {};
  for (int k0 = 0; k0 < IN_DIM; k0 += 4) {
    const int kk = k0 + 2 * hi;
    v2f a, b;
    a.x = x[rowA * IN_DIM + kk];
    a.y = x[rowA * IN_DIM + kk + 1];
    b.x = Wx[colB * IN_DIM + kk];
    b.y = Wx[colB * IN_DIM + kk + 1];
    // (neg_a, A, neg_b, B, c_mod, C, reuse_a, reuse_b)
    c = __builtin_amdgcn_wmma_f32_16x16x4_f32(false, a, false, b,
                                              (short)0, c, false, false);
  }

  const int col = n0 + lm;
  const float bv = bx[col];
#pragma unroll
  for (int v = 0; v < 8; ++v) {
    xf[(size_t)(m0 + v + 8 * hi) * G4H + col] = c[v] + bv;
  }
}

// ---------------------------------------------------------------------------
// Kernel 2: persistent recurrent scan. One workgroup per batch element,
// 256 threads (8 wave32s); hebb[b] (H*H fp32 = 256KB) lives in LDS for all
// T steps. Thread k owns hidden unit k (c, gates, hebb column k).
// LDS hebb layout [j][k]: lanes hit distinct banks -> conflict-free.
// ---------------------------------------------------------------------------
__global__ __launch_bounds__(256)
void plastic_lstm_scan_kernel(const float* __restrict__ xf,
                              const float* __restrict__ WhTP,
                              const float* __restrict__ bh,
                              const float* __restrict__ w_mod,
                              const float* __restrict__ b_mod,
                              const float* __restrict__ w_fan,
                              const float* __restrict__ b_fan,
                              const float* __restrict__ alpha,
                              float* __restrict__ out_hs,
                              float* __restrict__ out_h,
                              float* __restrict__ out_c,
                              float* __restrict__ out_hebb) {
  extern __shared__ float smem[];
  float* hebb_sh = smem;                 // H*H = 65536 floats (256 KB)
  float* h_sh    = smem + HID * HID;     // H floats
  float* red     = h_sh + HID;           // 8 partials + 1 eta

  const int k = threadIdx.x;             // hidden index owned by this thread
  const int b = blockIdx.x;              // batch element

  // --- init state (deterministic; never rely on prior buffer contents) ---
  h_sh[k] = 0.0f;
  for (int j0 = 0; j0 < HID; ++j0) hebb_sh[j0 * HID + k] = 0.0f;
  float c = 0.0f;

  const float bh0 = bh[k];
  const float bh1 = bh[HID + k];
  const float bh2 = bh[2 * HID + k];
  const float bh3 = bh[3 * HID + k];
  const float wmk = w_mod[k];
  const float wfk = w_fan[k];
  const float bfk = b_fan[k];
  const float bm  = b_mod[0];

  const v4f* __restrict__ wp = (const v4f*)WhTP;  // [j][k] of float4 gates
  __syncthreads();

  for (int t = 0; t < T_STEPS; ++t) {
    const float* xfp = xf + (size_t)(t * BATCH + b) * G4H;
    if (t + 1 < T_STEPS) {
      // hide next step's xf fetch behind this step's ~256-iter compute
      __builtin_prefetch(xf + (size_t)((t + 1) * BATCH + b) * G4H + k, 0, 1);
    }
    float a0 = xfp[k]           + bh0;
    float a1 = xfp[HID + k]     + bh1;
    float a2 = xfp[2 * HID + k] + bh2;
    float a3 = xfp[3 * HID + k] + bh3;
    float ap = 0.0f;  // plast[k]

    // Fused pass over h-dimension: recurrent matvec + plasticity term.
    // Per iter: 1x global_load_b128 (gates), 1x global_load_b32 (alpha),
    //           1x ds broadcast (h), 1x ds load (hebb), 6 FMAs.
#pragma unroll 4
    for (int j0 = 0; j0 < HID; ++j0) {
      const float hv = h_sh[j0];
      const v4f w = wp[j0 * HID + k];
      a0 = fmaf(hv, w.x, a0);
      a1 = fmaf(hv, w.y, a1);
      a2 = fmaf(hv, w.z, a2);
      a3 = fmaf(hv, w.w, a3);
      ap = fmaf(hv * alpha[j0 * HID + k], hebb_sh[j0 * HID + k], ap);
    }

    const float fg = 1.0f / (1.0f + expf(-a0));
    const float ig = 1.0f / (1.0f + expf(-a1));
    const float og = 1.0f / (1.0f + expf(-a2));
    const float jg = tanhf(a3 + ap);
    c = fg * c + ig * jg;
    const float hn = og * tanhf(c);

    // eta = tanh(dot(c_new, w_mod) + b_mod): wave32 tree + cross-wave reduce.
    float v = c * wmk;
    for (int off = 16; off > 0; off >>= 1) v += __shfl_down(v, off, 32);
    if ((k & 31) == 0) red[k >> 5] = v;
    __syncthreads();
    if (k == 0) {
      float s = red[0] + red[1] + red[2] + red[3] +
                red[4] + red[5] + red[6] + red[7];
      red[8] = tanhf(s + bm);
    }
    __syncthreads();
    const float myeta = red[8] * wfk + bfk;
    const float mj = myeta * jg;   // delta[h,k] = myeta[k]*h_old[h]*j[k]

    // Hebbian update against OLD h (h_sh untouched until the barrier below).
#pragma unroll 4
    for (int j0 = 0; j0 < HID; ++j0) {
      const int idx = j0 * HID + k;
      float vv = fmaf(mj, h_sh[j0], hebb_sh[idx]);
      vv = fminf(CLIPV, fmaxf(-CLIPV, vv));
      hebb_sh[idx] = vv;
    }
    __syncthreads();           // everyone done reading old h
    h_sh[k] = hn;
    out_hs[(size_t)(t * BATCH + b) * HID + k] = hn;
    __syncthreads();           // new h visible before next step
  }

  out_h[b * HID + k] = h_sh[k];
  out_c[b * HID + k] = c;
  for (int j0 = 0; j0 < HID; ++j0)
    out_hebb[(size_t)b * HID * HID + (size_t)j0 * HID + k] =
        hebb_sh[j0 * HID + k];
}

// ---------------------------------------------------------------------------
extern "C" void kernel_launch(void* const* d_in, const int* in_sizes, int n_in,
                              void* d_out, int out_size, void* d_ws,
                              size_t ws_size, hipStream_t stream) {
  (void)in_sizes; (void)n_in; (void)out_size; (void)ws_size;

  const float* x     = (const float*)d_in[0];  // [T,B,I]
  const float* Wx    = (const float*)d_in[1];  // [4H,I]
  const float* bx    = (const float*)d_in[2];  // [4H]
  const float* Wh    = (const float*)d_in[3];  // [4H,H]
  const float* bh    = (const float*)d_in[4];  // [4H]
  const float* w_mod = (const float*)d_in[5];  // [1,H]
  const float* b_mod = (const float*)d_in[6];  // [1]
  const float* w_fan = (const float*)d_in[7];  // [H,1]
  const float* b_fan = (const float*)d_in[8];  // [H]
  const float* alpha = (const float*)d_in[9];  // [H,H]

  float* out = (float*)d_out;
  float* out_hs   = out;                                       // T*B*H
  float* out_h    = out_hs + (size_t)T_STEPS * BATCH * HID;    // B*H
  float* out_c    = out_h + (size_t)BATCH * HID;               // B*H
  float* out_hebb = out_c + (size_t)BATCH * HID;               // B*H*H

  float* xf   = (float*)d_ws;                                  // 8192*1024
  float* WhTP = xf + (size_t)T_STEPS * BATCH * G4H;            // 256*256*4

  // 0) gate-interleaved repack of Wh for float4 loads in the scan
  wh_repack_kernel<<<(G4H * HID) / 256, 256, 0, stream>>>(Wh, WhTP);

  // 1) xf = x @ Wx^T + bx (fp32 WMMA). 32768 tiles, 4 waves/block.
  xf_gemm_wmma_kernel<<<8192, dim3(32, 4), 0, stream>>>(x, Wx, bx, xf);

  // 2) persistent scan; hebb resident in 263 KB of dynamic LDS per WGP.
  const size_t shmem = (size_t)(HID * HID + HID + 16) * sizeof(float);
  (void)hipFuncSetAttribute(
      reinterpret_cast<const void*>(plastic_lstm_scan_kernel),
      hipFuncAttributeMaxDynamicSharedMemorySize, (int)shmem);
  plastic_lstm_scan_kernel<<<BATCH, HID, shmem, stream>>>(
      xf, WhTP, bh, w_mod, b_mod, w_fan, b_fan, alpha,
      out_hs, out_h, out_c, out_hebb);
}